// SpatialGraphConv_49323404427949
// MI455X (gfx1250) — compile-verified
//
#include <hip/hip_runtime.h>

#define CIN   64
#define COUT  64
#define TT    12

typedef __attribute__((ext_vector_type(2))) float v2f;
typedef __attribute__((ext_vector_type(8))) float v8f;

// ---------------- graph-normalization precompute ----------------

// deg starts at 1.0 (the self-loop weight in PyG GCNConv default)
__global__ void k_deg_init(float* __restrict__ deg, int n) {
  int i = blockIdx.x * blockDim.x + threadIdx.x;
  if (i < n) deg[i] = 1.0f;
}

__global__ void k_deg_acc(const int* __restrict__ dst,
                          const float* __restrict__ ew,
                          float* __restrict__ deg, int e) {
  int i = blockIdx.x * blockDim.x + threadIdx.x;
  if (i < e) atomicAdd(&deg[dst[i]], ew[i]);
}

// in-place: deg -> d^{-1/2}
__global__ void k_dis(float* __restrict__ deg, int n) {
  int i = blockIdx.x * blockDim.x + threadIdx.x;
  if (i < n) {
    float d = deg[i];
    deg[i] = (d > 0.0f) ? rsqrtf(d) : 0.0f;
  }
}

// norm[e] = dis[src]*ew*dis[dst] ; normSelf[n] = dis[n]^2
__global__ void k_norm(const int* __restrict__ src, const int* __restrict__ dst,
                       const float* __restrict__ ew, const float* __restrict__ dis,
                       float* __restrict__ norm, float* __restrict__ normSelf,
                       int e, int n) {
  int i = blockIdx.x * blockDim.x + threadIdx.x;
  if (i < e) norm[i] = dis[src[i]] * ew[i] * dis[dst[i]];
  if (i < n) {
    float d = dis[i];
    normSelf[i] = d * d;
  }
}

// ---------------- per-slice WMMA GEMM: xw = x[:,:,t] @ W ----------------
// Also initializes acc = normSelf * xw  (fuses zero-init + self-loop edge).
// One wave -> 16 nodes x 64 couts; 16 K-steps x 4 cout tiles of
// v_wmma_f32_16x16x4_f32. W is staged in LDS in a K-pair-interleaved,
// bank-padded layout so each B fragment is one aligned ds_load_b64 landing
// directly in an even VGPR pair (no repacking movs before the WMMA).
//
// LDS layout: Wl[kpair * 160 + col * 2 + (k & 1)]
//   - kpair = k >> 1  (32 pairs)
//   - stride 160 DWORDs: lanes 0-15 (kpair even half) use banks 0..31,
//     lanes 16-31 (kpair odd half) use banks 32..63 -> conflict-free b64.
#define WL_STRIDE 160

__global__ void k_gemm(const float* __restrict__ x, const float* __restrict__ W,
                       const float* __restrict__ normSelf,
                       float* __restrict__ xw, float* __restrict__ acc,
                       int n, int t) {
  __shared__ float Wl[32 * WL_STRIDE];   // 20 KB
  for (int i = threadIdx.x; i < CIN * COUT; i += blockDim.x) {
    const int k = i >> 6;        // CIN index 0..63
    const int col = i & 63;      // COUT index
    Wl[(k >> 1) * WL_STRIDE + col * 2 + (k & 1)] = W[i];
  }
  __syncthreads();

  const int wave = threadIdx.x >> 5;      // wave32: 8 waves / 256-thread block
  const int lane = threadIdx.x & 31;
  const int nodeBase = (blockIdx.x * 8 + wave) * 16;
  if (nodeBase >= n) return;              // wave-uniform: EXEC stays all-1s

  const int row  = lane & 15;             // M (A) / N (B,C,D) index
  const int half = lane >> 4;             // lane half selects K pair
  int nd = nodeBase + row;
  if (nd > n - 1) nd = n - 1;             // clamp (address only, not EXEC)
  const float* __restrict__ xrow = x + (size_t)nd * (CIN * TT) + t;

  v8f c0 = {}, c1 = {}, c2 = {}, c3 = {};

#pragma unroll
  for (int ks = 0; ks < 16; ++ks) {
    const int k0 = ks * 4 + half * 2;     // this lane's two K values: k0, k0+1
    v2f a;
    a.x = xrow[(size_t)k0 * TT];          // x[nd][k0][t]
    a.y = xrow[(size_t)(k0 + 1) * TT];

    // B fragments: one contiguous 8B LDS load per cout tile
    const int kp = ks * 2 + half;         // = k0 >> 1
    const float* wp = &Wl[kp * WL_STRIDE + row * 2];
    v2f b0 = *(const v2f*)(wp +  0);      // cols  0..15
    v2f b1 = *(const v2f*)(wp + 32);      // cols 16..31
    v2f b2 = *(const v2f*)(wp + 64);      // cols 32..47
    v2f b3 = *(const v2f*)(wp + 96);      // cols 48..63

    c0 = __builtin_amdgcn_wmma_f32_16x16x4_f32(false, a, false, b0, (short)0, c0, false, false);
    c1 = __builtin_amdgcn_wmma_f32_16x16x4_f32(false, a, false, b1, (short)0, c1, false, false);
    c2 = __builtin_amdgcn_wmma_f32_16x16x4_f32(false, a, false, b2, (short)0, c2, false, false);
    c3 = __builtin_amdgcn_wmma_f32_16x16x4_f32(false, a, false, b3, (short)0, c3, false, false);
  }

  // C/D layout: VGPR v holds rows (v) [lanes 0-15] and (v+8) [lanes 16-31]
#pragma unroll
  for (int v = 0; v < 8; ++v) {
    const int ndOut = nodeBase + v + half * 8;
    if (ndOut < n) {
      const float ns = normSelf[ndOut];
      const size_t base = (size_t)ndOut * COUT + row;
      float y0 = c0[v], y1 = c1[v], y2 = c2[v], y3 = c3[v];
      xw[base +  0] = y0;  acc[base +  0] = ns * y0;
      xw[base + 16] = y1;  acc[base + 16] = ns * y1;
      xw[base + 32] = y2;  acc[base + 32] = ns * y2;
      xw[base + 48] = y3;  acc[base + 48] = ns * y3;
    }
  }
}

// ---------------- edge scatter: acc[dst] += norm[e] * xw[src] ----------------
// 64 threads per edge: coalesced gather + coalesced atomics on the compact,
// L2-resident acc buffer (25.6 MB working set << 192 MB L2).
__global__ void k_scatter(const int* __restrict__ src, const int* __restrict__ dst,
                          const float* __restrict__ norm,
                          const float* __restrict__ xw, float* __restrict__ acc,
                          int e) {
  const int tid = blockIdx.x * blockDim.x + threadIdx.x;
  const int eIdx = tid >> 6;
  const int c = tid & 63;
  if (eIdx >= e) return;
  const int s = src[eIdx];
  const int d = dst[eIdx];
  const float w = norm[eIdx];
  const float* srow = xw + (size_t)s * COUT;
  __builtin_prefetch(srow, 0, 1);     // global_prefetch_b8
  atomicAdd(&acc[(size_t)d * COUT + c], w * srow[c]);
}

// ---------------- epilogue: out[n, c, t] = relu(acc + b) ----------------
__global__ void k_writeout(const float* __restrict__ acc, const float* __restrict__ b,
                           float* __restrict__ out, int n, int t) {
  const int i = blockIdx.x * blockDim.x + threadIdx.x;
  if (i >= n * COUT) return;
  const int nd = i / COUT;
  const int c = i & 63;
  float v = acc[i] + b[c];
  out[(size_t)nd * (COUT * TT) + (size_t)c * TT + t] = (v > 0.0f) ? v : 0.0f;
}

extern "C" void kernel_launch(void* const* d_in, const int* in_sizes, int n_in,
                              void* d_out, int out_size, void* d_ws, size_t ws_size,
                              hipStream_t stream) {
  const float* x    = (const float*)d_in[0];  // (N, CIN, T)
  const int*   ei   = (const int*)  d_in[1];  // (2, E)
  const float* ea   = (const float*)d_in[2];  // (E,)
  const float* W    = (const float*)d_in[3];  // (CIN, COUT)
  const float* bias = (const float*)d_in[4];  // (COUT,)

  const int E = in_sizes[2];
  const int N = in_sizes[0] / (CIN * TT);
  const int* src = ei;
  const int* dst = ei + E;

  // workspace layout (floats): dis[N] | normSelf[N] | norm[E] | xw[N*64] | acc[N*64]
  float* ws       = (float*)d_ws;
  float* dis      = ws;
  float* normSelf = dis + N;
  float* norm     = normSelf + N;
  float* xw       = norm + E;
  float* acc      = xw + (size_t)N * COUT;

  float* out = (float*)d_out;

  const int mx = (E > N) ? E : N;
  k_deg_init<<<(N + 255) / 256, 256, 0, stream>>>(dis, N);
  k_deg_acc <<<(E + 255) / 256, 256, 0, stream>>>(dst, ea, dis, E);
  k_dis     <<<(N + 255) / 256, 256, 0, stream>>>(dis, N);
  k_norm    <<<(mx + 255) / 256, 256, 0, stream>>>(src, dst, ea, dis, norm, normSelf, E, N);

  const int gemmBlocks = (N + 127) / 128;                       // 8 waves x 16 nodes
  const long long scat = (long long)E * COUT;
  const int scatBlocks = (int)((scat + 255) / 256);
  const int woBlocks   = (N * COUT + 255) / 256;

  for (int t = 0; t < TT; ++t) {
    k_gemm    <<<gemmBlocks, 256, 0, stream>>>(x, W, normSelf, xw, acc, N, t);
    k_scatter <<<scatBlocks, 256, 0, stream>>>(src, dst, norm, xw, acc, E);
    k_writeout<<<woBlocks,   256, 0, stream>>>(acc, bias, out, N, t);
  }
}